// CrossLayerAttention_85572928405669
// MI455X (gfx1250) — compile-verified
//
#include <hip/hip_runtime.h>
#include <math.h>

typedef __attribute__((ext_vector_type(2))) float v2f;
typedef __attribute__((ext_vector_type(4))) float f4;
typedef __attribute__((ext_vector_type(8))) float v8f;

// ---------------------------------------------------------------------------
// Kernel 1: M = Wq^T * Wk  (64x64), written directly in WMMA B-fragment order:
//   fragment (nt, kc): lane holds (b0,b1) = M[4kc+koff+0][16nt+lane%16],
//                                 M[4kc+koff+1][...],  koff = 2*(lane>=16)
//   stored as v2f at index (nt*16+kc)*32 + lane  -> coalesced b64 loads later.
// ---------------------------------------------------------------------------
__global__ __launch_bounds__(64) void build_Mfrag(const float* __restrict__ Wq,
                                                  const float* __restrict__ Wk,
                                                  float* __restrict__ Mfrag) {
  const int d  = blockIdx.x;    // row of M (0..63)
  const int dp = threadIdx.x;   // col of M (0..63)
  float sum = 0.f;
#pragma unroll 8
  for (int e = 0; e < 64; ++e)
    sum = fmaf(Wq[e * 64 + d], Wk[e * 64 + dp], sum);
  const int kc   = d >> 2;
  const int vreg = d & 1;
  const int lh   = (d >> 1) & 1;              // lane half (K=0,1 vs K=2,3)
  const int nt   = dp >> 4;
  const int lane = lh * 16 + (dp & 15);
  Mfrag[(((nt * 16 + kc) * 32) + lane) * 2 + vreg] = sum;
}

// ---------------------------------------------------------------------------
// Kernel 2: fused cross-layer attention, one wave per 16-token tile.
//   Phase A: G(16x64) = X_tile(16x64) @ M(64x64) via v_wmma_f32_16x16x4_f32
//   Phase B: single streaming pass over L layers with online softmax:
//            out = sum_l e_l*s_l^2*x_l / (Z1 + 1e-6*Z0)
// ---------------------------------------------------------------------------
__global__ __launch_bounds__(256) void cla_fused(
    const float* __restrict__ cur, const float* __restrict__ all,
    const float* __restrict__ Mfrag, const float* __restrict__ scales,
    const float* __restrict__ temperature, float* __restrict__ out,
    int N, int L, int ntiles) {
  __shared__ __align__(16) float gsm[8][16 * 68];  // padded stride 68

  const int lane = threadIdx.x & 31;
  const int wid  = threadIdx.x >> 5;
  const int tile = blockIdx.x * 8 + wid;
  if (tile >= ntiles) return;   // wave-uniform; EXEC stays all-ones below
  const int n0 = tile * 16;

  // ---- Phase A: load A fragments (lane L: row L%16, K cols koff..koff+1 of
  //      each 4-wide K chunk) ----
  const int row  = lane & 15;
  const int koff = (lane >> 4) << 1;  // 0 or 2
  float a0[16], a1[16];
  {
    const float* xr = cur + (size_t)(n0 + row) * 64 + koff;
#pragma unroll
    for (int kc = 0; kc < 16; ++kc) {
      v2f v = *(const v2f*)(xr + 4 * kc);
      a0[kc] = v.x;
      a1[kc] = v.y;
    }
  }

  // ---- G = X @ M with fp32 WMMA; stage D fragments into LDS ----
  const v2f* mfb = (const v2f*)Mfrag;
#pragma unroll
  for (int nt = 0; nt < 4; ++nt) {
    v8f c = {0.f, 0.f, 0.f, 0.f, 0.f, 0.f, 0.f, 0.f};
#pragma unroll
    for (int kc = 0; kc < 16; ++kc) {
      v2f b = mfb[(nt * 16 + kc) * 32 + lane];
      v2f a = {a0[kc], a1[kc]};
      c = __builtin_amdgcn_wmma_f32_16x16x4_f32(
          /*neg_a=*/false, a, /*neg_b=*/false, b,
          /*c_mod=*/(short)0, c, /*reuse_a=*/false, /*reuse_b=*/false);
    }
    const int tokbase = (lane >> 4) << 3;   // D layout: vgpr j -> row j or j+8
    const int col     = nt * 16 + (lane & 15);
#pragma unroll
    for (int j = 0; j < 8; ++j)
      gsm[wid][(tokbase + j) * 68 + col] = c[j];
  }

  // ---- re-shard g: 2 lanes per token, 32 components each ----
  const int tok  = lane >> 1;
  const int half = lane & 1;
  float g[32];
  {
    const f4* gp = (const f4*)&gsm[wid][tok * 68 + half * 32];
#pragma unroll
    for (int j = 0; j < 8; ++j) {
      f4 v = gp[j];
      g[4 * j + 0] = v.x; g[4 * j + 1] = v.y;
      g[4 * j + 2] = v.z; g[4 * j + 3] = v.w;
    }
  }

  // ---- Phase B: one streaming pass over layers, online softmax ----
  const float invs = 1.0f / (8.0f * fabsf(temperature[0]));  // sqrt(64)=8
  float mrun = -3.0e38f, Z0 = 0.f, Z1 = 0.f;
  float acc[32];
#pragma unroll
  for (int j = 0; j < 32; ++j) acc[j] = 0.f;

  const float*  xbase   = all + (size_t)(n0 + tok) * 64 + half * 32;
  const size_t  lstride = (size_t)N * 64;

  for (int l = 0; l < L; ++l) {
    const f4* xp = (const f4*)(xbase + (size_t)l * lstride);
    float x[32];
#pragma unroll
    for (int j = 0; j < 8; ++j) {
      f4 v = __builtin_nontemporal_load(xp + j);  // streamed once: NT hint
      x[4 * j + 0] = v.x; x[4 * j + 1] = v.y;
      x[4 * j + 2] = v.z; x[4 * j + 3] = v.w;
    }
    float d = 0.f;
#pragma unroll
    for (int j = 0; j < 32; ++j) d = fmaf(g[j], x[j], d);
    d += __shfl_xor(d, 1, 32);                 // combine the two half-dots
    const float sc = d * invs;
    const float s  = scales[l];                // uniform -> s_load
    const float nm = fmaxf(mrun, sc);
    const float cf = __expf(mrun - nm);        // first iter: exp(-huge)=0
    const float p  = __expf(sc - nm);
    const float w  = p * s * s;
    Z0 = Z0 * cf + p;
    Z1 = Z1 * cf + p * s;
#pragma unroll
    for (int j = 0; j < 32; ++j) acc[j] = fmaf(acc[j], cf, w * x[j]);
    mrun = nm;
  }

  const float r = 1.0f / (Z1 + 1e-6f * Z0);
  float* op = out + (size_t)(n0 + tok) * 64 + half * 32;
#pragma unroll
  for (int j = 0; j < 8; ++j) {
    f4 v = {acc[4 * j + 0] * r, acc[4 * j + 1] * r,
            acc[4 * j + 2] * r, acc[4 * j + 3] * r};
    __builtin_nontemporal_store(v, (f4*)op + j);
  }
}

// ---------------------------------------------------------------------------
extern "C" void kernel_launch(void* const* d_in, const int* in_sizes, int n_in,
                              void* d_out, int out_size, void* d_ws,
                              size_t ws_size, hipStream_t stream) {
  const float* cur    = (const float*)d_in[0];  // [B,T,H,64]
  const float* all    = (const float*)d_in[1];  // [L,B,T,H,64]
  const float* Wq     = (const float*)d_in[2];  // [64,64]
  const float* Wk     = (const float*)d_in[3];  // [64,64]
  const float* scales = (const float*)d_in[4];  // [L]
  const float* temp   = (const float*)d_in[5];  // [1]
  // d_in[6] = current_layer_idx: unused by the math
  float* out   = (float*)d_out;
  float* Mfrag = (float*)d_ws;                  // 4096 floats = 16 KB

  const int N      = in_sizes[0] / 64;  // B*T*H token rows
  const int L      = in_sizes[4];       // number of layers
  const int ntiles = N / 16;

  build_Mfrag<<<64, 64, 0, stream>>>(Wq, Wk, Mfrag);

  const int blocks = (ntiles + 7) / 8;  // 8 waves (tiles) per 256-thread block
  cla_fused<<<blocks, 256, 0, stream>>>(cur, all, Mfrag, scales, temp, out, N,
                                        L, ntiles);
}